// CascadeUBBRROIHeads_20005957665009
// MI455X (gfx1250) — compile-verified
//
#include <hip/hip_runtime.h>
#include <cstdint>
#include <cstddef>

#define N_IMG       16
#define M_GT        128
#define P_PROP      8192
#define NUM_CLASSES 80
#define THRESH_LO   0.3f
#define THRESH_HI   0.7f
#define BLOCK       256

// One workgroup = one image (blockIdx.y) x one tile of 256 proposals (blockIdx.x).
// GT tile (boxes/classes/ignores, 3KB) staged into LDS via CDNA5 async-to-LDS DMA.
__global__ __launch_bounds__(BLOCK) void iou_match_kernel(
    const float* __restrict__ gt_boxes,    // [N, M, 4]
    const float* __restrict__ prop_boxes,  // [N, P, 4]
    const int*   __restrict__ gt_classes,  // [N, M]
    const int*   __restrict__ gt_ignores,  // [N, M] (bool as int32)
    float*       __restrict__ out)         // iou[N,M,P] ++ idx[N,P] ++ lab[N,P] ++ cls[N,P]
{
    __shared__ float4 s_box[M_GT];  // 2048 B
    __shared__ int    s_cls[M_GT];  //  512 B
    __shared__ int    s_ign[M_GT];  //  512 B

    const int t = threadIdx.x;
    const int n = blockIdx.y;
    const int p = blockIdx.x * BLOCK + t;

    // Prefetch this thread's proposal box while the GT tile DMA is in flight.
    const float4* pb_ptr = ((const float4*)prop_boxes) + (size_t)n * P_PROP + p;
    __builtin_prefetch(pb_ptr, 0, 3);  // global_prefetch_b8

    // ---- Stage per-image GT data into LDS with async-to-LDS DMA (ASYNCcnt) ----
    {
        // 256 lanes x 8B = 2048B of gt boxes. EXEC all-ones, per-lane addresses.
        uint32_t lds_box = (uint32_t)(size_t)(&s_box[0]) + (uint32_t)t * 8u;
        uint64_t ga_box  = (uint64_t)(size_t)((const char*)(gt_boxes + (size_t)n * M_GT * 4)
                                              + (size_t)t * 8);
        asm volatile("global_load_async_to_lds_b64 %0, %1, off"
                     :: "v"(lds_box), "v"(ga_box) : "memory");

        // Lanes 0..127 -> classes (512B), lanes 128..255 -> ignore flags (512B).
        uint32_t lds_ci;
        uint64_t ga_ci;
        if (t < M_GT) {
            lds_ci = (uint32_t)(size_t)(&s_cls[0]) + (uint32_t)t * 4u;
            ga_ci  = (uint64_t)(size_t)(gt_classes + (size_t)n * M_GT + t);
        } else {
            lds_ci = (uint32_t)(size_t)(&s_ign[0]) + (uint32_t)(t - M_GT) * 4u;
            ga_ci  = (uint64_t)(size_t)(gt_ignores + (size_t)n * M_GT + (t - M_GT));
        }
        asm volatile("global_load_async_to_lds_b32 %0, %1, off"
                     :: "v"(lds_ci), "v"(ga_ci) : "memory");

        asm volatile("s_wait_asynccnt 0" ::: "memory");
    }
    __syncthreads();

    // ---- Per-proposal IoU row + matcher reduction ----
    const float4 pb     = *pb_ptr;                       // global_load_b128, coalesced
    const float  area_p = (pb.z - pb.x) * (pb.w - pb.y);

    float maxv  = -1.0f;      // argmax over m, first-occurrence tie-break (strict >)
    int   maxi  = 0;
    float maxig = -1e30f;     // max IoU over ignored GTs

    float* iou_out = out + (size_t)n * M_GT * P_PROP + p;

    #pragma unroll 4
    for (int m = 0; m < M_GT; ++m) {
        const float4 gb     = s_box[m];                  // ds_load_b128 broadcast
        const float  area_g = (gb.z - gb.x) * (gb.w - gb.y);
        const float  iw     = fmaxf(fminf(pb.z, gb.z) - fmaxf(pb.x, gb.x), 0.0f);
        const float  ih     = fmaxf(fminf(pb.w, gb.w) - fmaxf(pb.y, gb.y), 0.0f);
        const float  inter  = iw * ih;
        const float  uni    = area_g + area_p - inter;
        const float  r      = inter * __builtin_amdgcn_rcpf(fmaxf(uni, 1e-9f));
        const float  iou    = (inter > 0.0f) ? r : 0.0f;

        iou_out[(size_t)m * P_PROP] = iou;               // lanes contiguous in p

        if (iou > maxv) { maxv = iou; maxi = m; }
        if (s_ign[m])   { maxig = fmaxf(maxig, iou); }
    }

    int label = (maxv >= THRESH_HI) ? 1 : ((maxv >= THRESH_LO) ? -1 : 0);
    if (maxig > THRESH_HI) label = -1;

    int cls = s_cls[maxi];
    if (label == 0)  cls = NUM_CLASSES;
    if (label == -1) cls = -1;

    float* o_idx = out + (size_t)N_IMG * M_GT * P_PROP;
    float* o_lab = o_idx + (size_t)N_IMG * P_PROP;
    float* o_cls = o_lab + (size_t)N_IMG * P_PROP;
    const size_t np = (size_t)n * P_PROP + p;
    o_idx[np] = (float)maxi;
    o_lab[np] = (float)label;
    o_cls[np] = (float)cls;
}

extern "C" void kernel_launch(void* const* d_in, const int* in_sizes, int n_in,
                              void* d_out, int out_size, void* d_ws, size_t ws_size,
                              hipStream_t stream) {
    (void)in_sizes; (void)n_in; (void)out_size; (void)d_ws; (void)ws_size;
    const float* gt_boxes   = (const float*)d_in[0];
    const float* prop_boxes = (const float*)d_in[1];
    const int*   gt_classes = (const int*)d_in[2];
    const int*   gt_ignores = (const int*)d_in[3];
    float*       out        = (float*)d_out;

    dim3 grid(P_PROP / BLOCK, N_IMG);
    iou_match_kernel<<<grid, BLOCK, 0, stream>>>(gt_boxes, prop_boxes,
                                                 gt_classes, gt_ignores, out);
}